// ProteinGIN_50972671869196
// MI455X (gfx1250) — compile-verified
//
#include <hip/hip_runtime.h>

typedef __attribute__((ext_vector_type(2))) float v2f;
typedef __attribute__((ext_vector_type(8))) float v8f;

#define N_NODES 65536
#define FDIM 256
#define HDIM 512

// ---------------------------------------------------------------------------
// Seed h with node_feats (h starts as feats so the scatter-add yields feats+agg)
// ---------------------------------------------------------------------------
__global__ __launch_bounds__(256) void gin_seed_kernel(const float4* __restrict__ in,
                                                       float4* __restrict__ out, int n4)
{
    int i = blockIdx.x * blockDim.x + threadIdx.x;
    if (i < n4) out[i] = in[i];
}

// ---------------------------------------------------------------------------
// Edge scatter: one wave32 per edge; lane l handles floats 4l..4l+3 and 128+4l..128+4l+3.
// feats/agg are L2-resident (64 MB each vs 192 MB L2): atomics hit L2.
// ---------------------------------------------------------------------------
__global__ __launch_bounds__(256) void gin_scatter_kernel(const float* __restrict__ feats,
                                                          const int* __restrict__ src,
                                                          const int* __restrict__ dst,
                                                          float* __restrict__ h, int E)
{
    int wave = blockIdx.x * 8 + (threadIdx.x >> 5);
    int lane = threadIdx.x & 31;
    if (wave >= E) return;
    int s = src[wave];
    int d = dst[wave];
    const float4* fs = (const float4*)(feats + (size_t)s * FDIM) + lane;
    float* hb = h + (size_t)d * FDIM + lane * 4;
#pragma unroll
    for (int half = 0; half < 2; ++half) {
        float4 v = fs[half * 32];
        float* p = hb + half * 128;
        unsafeAtomicAdd(p + 0, v.x);
        unsafeAtomicAdd(p + 1, v.y);
        unsafeAtomicAdd(p + 2, v.z);
        unsafeAtomicAdd(p + 3, v.w);
    }
}

// ---------------------------------------------------------------------------
// f32 GEMM + bias + ReLU via V_WMMA_F32_16X16X4_F32 (exact f32 numerics).
// Wave tile 32(M)x32(N) = 4 accumulators; block = 8 waves -> 64x128 tile.
//
// A-fragment (16x4 f32): lane<16 -> row=lm, K=k0..k0+1 ; lane>=16 -> K=k0+2..k0+3
//   -> per-lane contiguous v2f load at A[row*K + k0 + lh*2].
// B-fragment (4x16 f32): VGPR0 holds K=(k0+lh*2) row, VGPR1 holds K=(k0+lh*2+1),
//   col = nBase + lm  -> two b32 loads, N*4 bytes apart.
// C/D (8 VGPRs): VGPR r -> row r (lanes 0-15) / row r+8 (lanes 16-31), col = lm.
// ---------------------------------------------------------------------------
template<int K, int N>
__global__ __launch_bounds__(256) void gin_gemm_relu_kernel(const float* __restrict__ A,
                                                            const float* __restrict__ B,
                                                            const float* __restrict__ bias,
                                                            float* __restrict__ C)
{
    const int lane = threadIdx.x & 31;
    const int wave = threadIdx.x >> 5;
    const int wm = wave & 1;        // 2 M-waves
    const int wn = wave >> 1;       // 4 N-waves
    const int lm = lane & 15;
    const int lh = lane >> 4;

    const int mBase = blockIdx.x * 64 + wm * 32;
    const int nBase = blockIdx.y * 128 + wn * 32;

    v8f acc00 = {}, acc01 = {}, acc10 = {}, acc11 = {};

    const float* a0p = A + (size_t)(mBase + lm) * K + lh * 2;
    const float* a1p = a0p + (size_t)16 * K;
    const float* bp0 = B + (size_t)(lh * 2) * N + nBase + lm;

#pragma unroll 4
    for (int k0 = 0; k0 < K; k0 += 4) {
        v2f a0 = *(const v2f*)(a0p + k0);
        v2f a1 = *(const v2f*)(a1p + k0);
        const float* bp = bp0 + (size_t)k0 * N;
        v2f b0, b1;
        b0[0] = bp[0];
        b0[1] = bp[N];
        b1[0] = bp[16];
        b1[1] = bp[N + 16];
        acc00 = __builtin_amdgcn_wmma_f32_16x16x4_f32(false, a0, false, b0, (short)0, acc00, false, false);
        acc01 = __builtin_amdgcn_wmma_f32_16x16x4_f32(false, a0, false, b1, (short)0, acc01, false, false);
        acc10 = __builtin_amdgcn_wmma_f32_16x16x4_f32(false, a1, false, b0, (short)0, acc10, false, false);
        acc11 = __builtin_amdgcn_wmma_f32_16x16x4_f32(false, a1, false, b1, (short)0, acc11, false, false);
    }

    // Epilogue: bias + ReLU, fused from the C-fragment layout.
    auto store_tile = [&](const v8f& acc, int t, int s) {
        const int col = nBase + s * 16 + lm;
        const float bv = bias[col];
        float* out = C + (size_t)(mBase + t * 16 + lh * 8) * N + col;
#pragma unroll
        for (int r = 0; r < 8; ++r) {
            float v = acc[r] + bv;
            out[(size_t)r * N] = v > 0.0f ? v : 0.0f;
        }
    };
    store_tile(acc00, 0, 0);
    store_tile(acc01, 0, 1);
    store_tile(acc10, 1, 0);
    store_tile(acc11, 1, 1);
}

// ---------------------------------------------------------------------------
// Inputs (setup_inputs order): node_feats, src, dst, W1, b1, W2, b2, batch_size
// d_ws layout: h (65536*256 f32, 64 MB) | h1 (65536*512 f32, 128 MB)
// ---------------------------------------------------------------------------
extern "C" void kernel_launch(void* const* d_in, const int* in_sizes, int n_in,
                              void* d_out, int out_size, void* d_ws, size_t ws_size,
                              hipStream_t stream) {
    const float* node_feats = (const float*)d_in[0];
    const int*   src        = (const int*)d_in[1];
    const int*   dst        = (const int*)d_in[2];
    const float* W1         = (const float*)d_in[3];
    const float* b1         = (const float*)d_in[4];
    const float* W2         = (const float*)d_in[5];
    const float* b2         = (const float*)d_in[6];
    const int E = in_sizes[1];

    float* h   = (float*)d_ws;
    float* h1  = h + (size_t)N_NODES * FDIM;
    float* out = (float*)d_out;

    // 1) h = node_feats
    {
        int n4 = N_NODES * FDIM / 4;
        gin_seed_kernel<<<n4 / 256, 256, 0, stream>>>((const float4*)node_feats, (float4*)h, n4);
    }

    // 2) h += segment_sum(feats[src], dst)   (one wave per edge)
    gin_scatter_kernel<<<(E + 7) / 8, 256, 0, stream>>>(node_feats, src, dst, h, E);

    // 3) h1 = relu(h @ W1 + b1)
    dim3 grid(N_NODES / 64, HDIM / 128);
    gin_gemm_relu_kernel<FDIM, HDIM><<<grid, 256, 0, stream>>>(h, W1, b1, h1);

    // 4) out = relu(h1 @ W2 + b2)
    gin_gemm_relu_kernel<HDIM, HDIM><<<grid, 256, 0, stream>>>(h1, W2, b2, out);
}